// GNN_63333587746840
// MI455X (gfx1250) — compile-verified
//
#include <hip/hip_runtime.h>
#include <hip/hip_bf16.h>
#include <math.h>
#include <stdint.h>

typedef __attribute__((ext_vector_type(2))) float v2f;
typedef __attribute__((ext_vector_type(8))) float v8f;
typedef __attribute__((ext_vector_type(4))) unsigned int u32x4;
typedef __attribute__((ext_vector_type(8))) int i32x8;
typedef __attribute__((ext_vector_type(4))) int i32x4;

#define HID     64
#define IN_CH   128
#define H4      256
#define PE_CFAC (-0.29710775379409f)   /* -ln(10000)/31 */
#define T_SCALE 31.25f                 /* RESCALE/T = 4000/128 */

#if defined(__has_builtin)
#if __has_builtin(__builtin_amdgcn_tensor_load_to_lds)
#define HAVE_TDM 1
#endif
#endif

// fp32 WMMA 16x16x4: D = A(16x4) * B(4x16) + C, full fp32 precision.
// Lane l: A elem = A[l&15][2*(l>>4)+v], B elem = B[2*(l>>4)+v][l&15],
// C/D elem v = [v + 8*(l>>4)][l&15].
static __device__ __forceinline__ v8f wmma4(v2f a, v2f b, v8f c) {
  return __builtin_amdgcn_wmma_f32_16x16x4_f32(false, a, false, b, (short)0, c,
                                               false, false);
}

static __device__ __forceinline__ void lds_fence_wave() {
  asm volatile("s_wait_dscnt 0x0" ::: "memory");
  __builtin_amdgcn_wave_barrier();
}

// ---------------------------------------------------------------------------
// Pre-pack a row-major weight W[Kd,Nd] into WMMA B-fragment order:
// out[(kstep*(Nd/16)+nt)*32 + lane] = { W[k][n], W[k+1][n] },
// k = 4*kstep + 2*(lane>>4), n = 16*nt + (lane&15).
// After this, every B fragment is ONE global_load_b64 / ds_load_b64.
// ---------------------------------------------------------------------------
__global__ void pack_weights_kernel(const float* __restrict__ W, int Kd, int Nd,
                                    v2f* __restrict__ out) {
  const int t = blockIdx.x * blockDim.x + threadIdx.x;
  const int ntiles = Nd >> 4;
  const int total = (Kd >> 2) * ntiles * 32;
  if (t >= total) return;
  const int l = t & 31;
  const int nt = (t >> 5) % ntiles;
  const int kstep = t / (32 * ntiles);
  const int k = kstep * 4 + 2 * (l >> 4);
  const int n = nt * 16 + (l & 15);
  v2f b;
  b.x = W[k * Nd + n];
  b.y = W[(k + 1) * Nd + n];
  out[t] = b;
}

// ---------------------------------------------------------------------------
// Embedder: h_embed = (x@W_in + b_in) + MLP_t(sinusoidal_pe(t))
// One wave per 16-row tile. All GEMMs on fp32 WMMA, B from packed fragments.
// ---------------------------------------------------------------------------
__global__ void embed_kernel(const float* __restrict__ x,
                             const int* __restrict__ tsteps,
                             const v2f* __restrict__ Winp,
                             const float* __restrict__ bin,
                             const v2f* __restrict__ Wt1p,
                             const float* __restrict__ bt1,
                             const v2f* __restrict__ Wt2p,
                             const float* __restrict__ bt2,
                             float* __restrict__ h_embed, int Nn) {
  __shared__ float uswap[8][16][17];  // per-wave 16x16 layout-swap tiles
  const int wid = threadIdx.x >> 5;
  const int l = threadIdx.x & 31;
  const int mrow = l & 15;
  const int khalf = l >> 4;
  const int m0 = (blockIdx.x * 8 + wid) * 16;
  if (m0 >= Nn) return;
  const int marow = min(m0 + mrow, Nn - 1);

  // ---- h = x @ W_in   (K = 128) ----
  v8f hacc[4] = {};
  for (int kk = 0; kk < IN_CH; kk += 4) {
    const int k = kk + 2 * khalf;
    v2f a = *(const v2f*)(x + (size_t)marow * IN_CH + k);
    const v2f* bp = Winp + ((kk >> 2) * 4) * 32 + l;
#pragma unroll
    for (int nt = 0; nt < 4; ++nt)
      hacc[nt] = wmma4(a, bp[nt * 32], hacc[nt]);
  }

  // ---- sinusoidal PE directly in A-fragment layout ----
  const float tval = (float)tsteps[marow] * T_SCALE;
  v2f peA[16];
#pragma unroll
  for (int i = 0; i < 16; ++i) {
    const int k = i * 4 + 2 * khalf;
#pragma unroll
    for (int v = 0; v < 2; ++v) {
      const int j = k + v;
      const float ang = tval * __expf((float)(j & 31) * PE_CFAC);
      const float pe = (j < 32) ? __sinf(ang) : __cosf(ang);
      if (v == 0) peA[i].x = pe; else peA[i].y = pe;
    }
  }

  // ---- t_emb = silu(pe@W_t1 + b_t1) @ W_t2 ----
  // 256-wide hidden processed one 16-col tile at a time; C/D->A layout swap
  // through wave-private LDS, guarded by s_wait_dscnt + wave barrier.
  v8f tacc[4] = {};
  for (int jt = 0; jt < 16; ++jt) {
    v8f u = {};
#pragma unroll
    for (int i = 0; i < 16; ++i)
      u = wmma4(peA[i], Wt1p[(i * 16 + jt) * 32 + l], u);
    const float bu = bt1[jt * 16 + mrow];
#pragma unroll
    for (int v = 0; v < 8; ++v) {
      const float z = u[v] + bu;
      u[v] = z * (1.0f / (1.0f + __expf(-z)));  // SiLU
    }
#pragma unroll
    for (int v = 0; v < 8; ++v)
      uswap[wid][v + 8 * khalf][mrow] = u[v];
    lds_fence_wave();
#pragma unroll
    for (int ks = 0; ks < 4; ++ks) {
      const int k2 = ks * 4 + 2 * khalf;
      v2f a2;
      a2.x = uswap[wid][mrow][k2];
      a2.y = uswap[wid][mrow][k2 + 1];
      const v2f* bp = Wt2p + ((jt * 4 + ks) * 4) * 32 + l;
#pragma unroll
      for (int nt = 0; nt < 4; ++nt)
        tacc[nt] = wmma4(a2, bp[nt * 32], tacc[nt]);
    }
    lds_fence_wave();  // WAR guard before next iteration overwrites the tile
  }

  // ---- h_embed = h + b_in + t_emb + b_t2 ----
#pragma unroll
  for (int nt = 0; nt < 4; ++nt) {
    const int n = nt * 16 + mrow;
    const float bb = bin[n] + bt2[n];
#pragma unroll
    for (int v = 0; v < 8; ++v) {
      const int m = m0 + v + 8 * khalf;
      if (m < Nn) h_embed[(size_t)m * HID + n] = hacc[nt][v] + tacc[nt][v] + bb;
    }
  }
}

// ---------------------------------------------------------------------------
// Y = act(A + abias) @ W   for A[Nn,64], W packed fragments of [64,64].
// Optional fused input bias+ReLU (folds the previous layer's epilogue into
// this kernel's A-read, saving a full [N,64] read+write pass).
// ---------------------------------------------------------------------------
__global__ void linear64_kernel(const float* __restrict__ A,
                                const v2f* __restrict__ Wp,
                                const float* __restrict__ abias,  // or nullptr
                                float* __restrict__ Y, int Nn) {
  const int wid = threadIdx.x >> 5;
  const int l = threadIdx.x & 31;
  const int mrow = l & 15;
  const int khalf = l >> 4;
  const int m0 = (blockIdx.x * 8 + wid) * 16;
  if (m0 >= Nn) return;
  const int marow = min(m0 + mrow, Nn - 1);

  v8f acc[4] = {};
  for (int kk = 0; kk < HID; kk += 4) {
    const int k = kk + 2 * khalf;
    v2f a = *(const v2f*)(A + (size_t)marow * HID + k);
    if (abias) {  // fused bias + ReLU on the input activation
      a.x = fmaxf(a.x + abias[k], 0.0f);
      a.y = fmaxf(a.y + abias[k + 1], 0.0f);
    }
    const v2f* bp = Wp + ((kk >> 2) * 4) * 32 + l;
#pragma unroll
    for (int nt = 0; nt < 4; ++nt)
      acc[nt] = wmma4(a, bp[nt * 32], acc[nt]);
  }
#pragma unroll
  for (int nt = 0; nt < 4; ++nt) {
    const int n = nt * 16 + mrow;
#pragma unroll
    for (int v = 0; v < 8; ++v) {
      const int m = m0 + v + 8 * khalf;
      if (m < Nn) Y[(size_t)m * HID + n] = acc[nt][v];
    }
  }
}

// ---------------------------------------------------------------------------
// Graph plumbing
// ---------------------------------------------------------------------------
__global__ void degree_kernel(const int* __restrict__ ei, float* __restrict__ deg,
                              int E) {
  const int t = blockIdx.x * blockDim.x + threadIdx.x;
  if (t < E) atomicAdd(&deg[ei[E + t]], 1.0f);  // col = ei[1][e]
}

__global__ void dinv_kernel(float* __restrict__ deg, int Nn) {
  const int t = blockIdx.x * blockDim.x + threadIdx.x;
  if (t < Nn) deg[t] = __frsqrt_rn(deg[t] + 1.0f);  // +1: self-loop
}

// out[col] += y[row] * dinv[row]*dinv[col], 16 lanes per edge, float4 grain.
__global__ void aggregate_kernel(const int* __restrict__ ei,
                                 const float* __restrict__ dinv,
                                 const float* __restrict__ Y,
                                 float* __restrict__ Out, int E, int Nn) {
  const long t = (long)blockIdx.x * blockDim.x + threadIdx.x;
  const long total = (long)(E + Nn) * 16;
  if (t >= total) return;
  const int e = (int)(t >> 4);
  const int c = ((int)t & 15) * 4;
  int r, cl;
  if (e < E) { r = ei[e]; cl = ei[E + e]; }
  else       { r = e - E; cl = r; }        // self-loop
  const float w = dinv[r] * dinv[cl];
  const float4 v = *(const float4*)(Y + (size_t)r * HID + c);
  float* o = Out + (size_t)cl * HID + c;
  atomicAdd(o + 0, v.x * w);
  atomicAdd(o + 1, v.y * w);
  atomicAdd(o + 2, v.z * w);
  atomicAdd(o + 3, v.w * w);
}

__global__ void bias_act_kernel(float* __restrict__ X, const float* __restrict__ b,
                                long total, int relu) {
  const long t = (long)blockIdx.x * blockDim.x + threadIdx.x;
  if (t >= total) return;
  float v = X[t] + b[(int)(t & (HID - 1))];
  if (relu) v = fmaxf(v, 0.0f);
  X[t] = v;
}

// ---------------------------------------------------------------------------
// Edge decoder. W_d1 (packed fragments, 32 KB) is staged into LDS by the
// Tensor Data Mover (one 2D descriptor: 8192 x f32 row), waited with
// s_wait_tensorcnt. One wave per 16 edges; fused SiLU + W_d2 dot +
// shfl_xor cross-lane reduction.
// ---------------------------------------------------------------------------
__global__ void decoder_kernel(const int* __restrict__ fei,
                               const float* __restrict__ henc,
                               const v2f* __restrict__ Wd1p,
                               const float* __restrict__ bd1,
                               const float* __restrict__ Wd2,
                               const float* __restrict__ bd2,
                               float* __restrict__ logits, int EF) {
  __shared__ v2f w1p[4096];  // 32 KB, packed B fragments; LDS offset 0

#ifdef HAVE_TDM
  if ((threadIdx.x >> 5) == 0) {  // wave 0 issues the TDM transfer
    const uint64_t ga = (uint64_t)(uintptr_t)Wd1p;
    u32x4 g0;
    g0.x = 1u;                                   // count=1 (valid descriptor)
    g0.y = 0u;                                   // lds_addr = 0 (w1p base)
    g0.z = (unsigned)(ga & 0xFFFFFFFFu);         // global_addr[31:0]
    g0.w = (unsigned)((ga >> 32) & 0x1FFFFFFu) | (2u << 30);  // addr hi | type=2
    // data_size=4B; tensor = 8192 x 1 f32; tile = 8192 x 1; stride = 8192
    i32x8 g1 = {0x00020000, 0x20000000, 0x00010000, 0x20000000,
                1, 8192, 0, 0};
    i32x4 gz4 = {0, 0, 0, 0};
    i32x8 gz8 = {0, 0, 0, 0, 0, 0, 0, 0};
    __builtin_amdgcn_tensor_load_to_lds(g0, g1, gz4, gz4, gz8, 0);
    __builtin_amdgcn_s_wait_tensorcnt(0);
  }
#else
  for (int i = threadIdx.x; i < 4096; i += blockDim.x) w1p[i] = Wd1p[i];
#endif
  __syncthreads();

  const int wid = threadIdx.x >> 5;
  const int l = threadIdx.x & 31;
  const int mrow = l & 15;
  const int khalf = l >> 4;
  const int e0 = (blockIdx.x * 8 + wid) * 16;
  if (e0 >= EF) return;
  const int ee = min(e0 + mrow, EF - 1);
  const int s = fei[ee];
  const int d = fei[EF + ee];

  v8f acc[4] = {};
  for (int kk = 0; kk < 2 * HID; kk += 4) {
    const int k = kk + 2 * khalf;
    // kk < HID is wave-uniform (k and kk differ by < 4)
    v2f a = (kk < HID) ? *(const v2f*)(henc + (size_t)s * HID + k)
                       : *(const v2f*)(henc + (size_t)d * HID + (k - HID));
    const v2f* bp = w1p + ((kk >> 2) * 4) * 32 + l;
#pragma unroll
    for (int nt = 0; nt < 4; ++nt)
      acc[nt] = wmma4(a, bp[nt * 32], acc[nt]);  // ds_load_b64 per fragment
  }

  // SiLU + dot with W_d2 (OUT=1): per-lane partials over owned columns
  float red[8];
#pragma unroll
  for (int v = 0; v < 8; ++v) red[v] = 0.0f;
#pragma unroll
  for (int nt = 0; nt < 4; ++nt) {
    const int n = nt * 16 + mrow;
    const float b1 = bd1[n];
    const float w2 = Wd2[n];
#pragma unroll
    for (int v = 0; v < 8; ++v) {
      const float z = acc[nt][v] + b1;
      red[v] += (z * (1.0f / (1.0f + __expf(-z)))) * w2;
    }
  }
#pragma unroll
  for (int off = 1; off < 16; off <<= 1) {
#pragma unroll
    for (int v = 0; v < 8; ++v) red[v] += __shfl_xor(red[v], off, 32);
  }
  if (mrow == 0) {
    const float b2 = bd2[0];
#pragma unroll
    for (int v = 0; v < 8; ++v) {
      const int e = e0 + khalf * 8 + v;
      if (e < EF) logits[e] = red[v] + b2;
    }
  }
}

// ---------------------------------------------------------------------------
extern "C" void kernel_launch(void* const* d_in, const int* in_sizes, int n_in,
                              void* d_out, int out_size, void* d_ws, size_t ws_size,
                              hipStream_t stream) {
  const float* x    = (const float*)d_in[0];
  const int*   ei   = (const int*)d_in[1];
  const int*   fei  = (const int*)d_in[2];
  const int*   ts   = (const int*)d_in[3];
  const float* Win  = (const float*)d_in[4];
  const float* bin  = (const float*)d_in[5];
  const float* Wt1  = (const float*)d_in[6];
  const float* bt1  = (const float*)d_in[7];
  const float* Wt2  = (const float*)d_in[8];
  const float* bt2  = (const float*)d_in[9];
  const float* Wc0  = (const float*)d_in[10];
  const float* bc0  = (const float*)d_in[11];
  const float* Wc1  = (const float*)d_in[12];
  const float* bc1  = (const float*)d_in[13];
  const float* Wd1  = (const float*)d_in[14];
  const float* bd1  = (const float*)d_in[15];
  const float* Wd2  = (const float*)d_in[16];
  const float* bd2  = (const float*)d_in[17];

  const int Nn = in_sizes[3];
  const int E  = in_sizes[1] / 2;
  const int EF = in_sizes[2] / 2;

  float* out     = (float*)d_out;
  float* logits  = out;                        // [EF]
  float* h_embed = out + EF;                   // [Nn*HID]
  float* h_enc   = h_embed + (size_t)Nn * HID; // [Nn*HID]

  float* dinv = (float*)d_ws;                  // [Nn]
  float* tmp1 = dinv + Nn;                     // [Nn*HID]  linear output
  float* tmp2 = tmp1 + (size_t)Nn * HID;       // [Nn*HID]  aggregation output
  v2f* Winp = (v2f*)(tmp2 + (size_t)Nn * HID); // packed fragment buffers
  v2f* Wt1p = Winp + 4096;                     // (K/4)*(N/16)*32 entries each
  v2f* Wt2p = Wt1p + 8192;
  v2f* Wc0p = Wt2p + 8192;
  v2f* Wc1p = Wc0p + 2048;
  v2f* Wd1p = Wc1p + 2048;                     // 4096 entries (32 KB)

  // 0) pack all weights into WMMA B-fragment order (tiny, one-time per call)
  pack_weights_kernel<<<(4096 + 255) / 256, 256, 0, stream>>>(Win, IN_CH, HID, Winp);
  pack_weights_kernel<<<(8192 + 255) / 256, 256, 0, stream>>>(Wt1, HID, H4, Wt1p);
  pack_weights_kernel<<<(8192 + 255) / 256, 256, 0, stream>>>(Wt2, H4, HID, Wt2p);
  pack_weights_kernel<<<(2048 + 255) / 256, 256, 0, stream>>>(Wc0, HID, HID, Wc0p);
  pack_weights_kernel<<<(2048 + 255) / 256, 256, 0, stream>>>(Wc1, HID, HID, Wc1p);
  pack_weights_kernel<<<(4096 + 255) / 256, 256, 0, stream>>>(Wd1, 2 * HID, HID, Wd1p);

  const int tilesN  = (Nn + 15) / 16;
  const int blocksN = (tilesN + 7) / 8;

  // 1) embedder -> h_embed (also an output)
  embed_kernel<<<blocksN, 256, 0, stream>>>(x, ts, Winp, bin, Wt1p, bt1, Wt2p,
                                            bt2, h_embed, Nn);

  // 2) degree + symmetric norm (self-loops folded as +1)
  (void)hipMemsetAsync(dinv, 0, sizeof(float) * Nn, stream);
  degree_kernel<<<(E + 255) / 256, 256, 0, stream>>>(ei, dinv, E);
  dinv_kernel<<<(Nn + 255) / 256, 256, 0, stream>>>(dinv, Nn);

  const long aggThreads = (long)(E + Nn) * 16;
  const int  aggBlocks  = (int)((aggThreads + 255) / 256);
  const long nh         = (long)Nn * HID;
  const int  nhBlocks   = (int)((nh + 255) / 256);

  // 3) GCN layer 0: linear -> aggregate   (epilogue fused into layer 1)
  linear64_kernel<<<blocksN, 256, 0, stream>>>(h_embed, Wc0p, nullptr, tmp1, Nn);
  (void)hipMemsetAsync(tmp2, 0, sizeof(float) * (size_t)Nn * HID, stream);
  aggregate_kernel<<<aggBlocks, 256, 0, stream>>>(ei, dinv, tmp1, tmp2, E, Nn);

  // 4) GCN layer 1: (bias+ReLU fused) linear -> aggregate -> +bias => h_enc
  linear64_kernel<<<blocksN, 256, 0, stream>>>(tmp2, Wc1p, bc0, tmp1, Nn);
  (void)hipMemsetAsync(h_enc, 0, sizeof(float) * (size_t)Nn * HID, stream);
  aggregate_kernel<<<aggBlocks, 256, 0, stream>>>(ei, dinv, tmp1, h_enc, E, Nn);
  bias_act_kernel<<<nhBlocks, 256, 0, stream>>>(h_enc, bc1, nh, 0);

  // 5) edge decoder -> logits (output)
  const int tilesE  = (EF + 15) / 16;
  const int blocksE = (tilesE + 7) / 8;
  decoder_kernel<<<blocksE, 256, 0, stream>>>(fei, h_enc, Wd1p, bd1, Wd2, bd2,
                                              logits, EF);
}